// GraphAttentionLayer_24670292148954
// MI455X (gfx1250) — compile-verified
//
#include <hip/hip_runtime.h>
#include <math.h>

typedef __attribute__((ext_vector_type(2))) float v2f;
typedef __attribute__((ext_vector_type(8))) float v8f;

#define IN_DIM    128
#define OUT_DIM   64
#define BB        8
#define NN        2048
#define NEG_SLOPE 0.2f

// ---------------------------------------------------------------------------
// Kernel 1: h = x @ W   (B*N = 16384 rows, K=128, N=64), fp32 WMMA 16x16x4.
// One wave computes a 16x64 tile of h. A-frag per ISA layout:
//   lanes 0-15 : row = r0+lane,   K = k0+0 / k0+1  (v0 / v1)
//   lanes16-31 : row = r0+lane-16,K = k0+2 / k0+3
// ---------------------------------------------------------------------------
__global__ __launch_bounds__(128) void gat_h_kernel(
    const float* __restrict__ x, const float* __restrict__ W,
    float* __restrict__ h) {
  const int wave = blockIdx.x * (blockDim.x >> 5) + (threadIdx.x >> 5);
  const int lane = threadIdx.x & 31;
  const int r0   = wave * 16;                 // tile of 16 flattened (b,n) rows
  const int mrow = lane & 15;
  const int koff = (lane >> 4) << 1;          // 0 or 2

  v8f acc[4] = {};                            // 4 n-subtiles of 16 cols
  const float* xrow = x + (size_t)(r0 + mrow) * IN_DIM;

  for (int k0 = 0; k0 < IN_DIM; k0 += 4) {
    v2f a;
    a.x = xrow[k0 + koff];
    a.y = xrow[k0 + koff + 1];
#pragma unroll
    for (int t = 0; t < 4; ++t) {
      const int n = t * 16 + mrow;
      v2f b;
      b.x = W[(k0 + koff)     * OUT_DIM + n];
      b.y = W[(k0 + koff + 1) * OUT_DIM + n];
      acc[t] = __builtin_amdgcn_wmma_f32_16x16x4_f32(
          false, a, false, b, (short)0, acc[t], false, false);
    }
  }
  // C/D layout: vgpr r, lanes 0-15 -> (M=r,   N=n0+lane)
  //                     lanes16-31 -> (M=r+8, N=n0+lane-16)
#pragma unroll
  for (int t = 0; t < 4; ++t)
#pragma unroll
    for (int r = 0; r < 8; ++r) {
      const int m = r + ((lane >> 4) << 3);
      h[(size_t)(r0 + m) * OUT_DIM + t * 16 + mrow] = acc[t][r];
    }
}

// ---------------------------------------------------------------------------
// Kernel 1b: f_i = h . a[:64], f_j = h . a[64:]   (one thread per row)
// ---------------------------------------------------------------------------
__global__ __launch_bounds__(256) void gat_f_kernel(
    const float* __restrict__ h, const float* __restrict__ a,
    float* __restrict__ fi, float* __restrict__ fj) {
  const int row = blockIdx.x * blockDim.x + threadIdx.x;   // < B*N
  const float4* hv = (const float4*)(h + (size_t)row * OUT_DIM);
  const float4* ai = (const float4*)a;
  const float4* aj = (const float4*)(a + OUT_DIM);
  float si = 0.f, sj = 0.f;
#pragma unroll
  for (int k = 0; k < OUT_DIM / 4; ++k) {
    const float4 hh = hv[k], vi = ai[k], vj = aj[k];
    si += hh.x * vi.x + hh.y * vi.y + hh.z * vi.z + hh.w * vi.w;
    sj += hh.x * vj.x + hh.y * vj.y + hh.z * vj.z + hh.w * vj.w;
  }
  fi[row] = si;
  fj[row] = sj;
}

// ---------------------------------------------------------------------------
// Kernel 2: per (batch, 16-row i-tile): masked leaky softmax -> alpha (global,
// coalesced), then out = alpha @ h via fp32 WMMA (alpha re-read L2-hot).
// 256 threads = 8 waves. Waves split the j dimension (256 cols each) for the
// matmul; partial 16x64 tiles reduced through LDS.
// ---------------------------------------------------------------------------
__global__ __launch_bounds__(256) void gat_attn_kernel(
    const float* __restrict__ adj, const float* __restrict__ h,
    const float* __restrict__ fi_g, const float* __restrict__ fj_g,
    float* __restrict__ out, float* __restrict__ alpha) {
  __shared__ float lds_fj[NN];
  __shared__ float lds_fi[16];
  __shared__ float lds_m[16];
  __shared__ float lds_is[16];
  __shared__ float lds_red[8 * 16 * OUT_DIM];   // 32 KB wave partials

  const int b  = blockIdx.x >> 7;               // NN/16 = 128 tiles per batch
  const int i0 = (blockIdx.x & 127) * 16;
  const int t  = threadIdx.x;

  for (int k = 0; k < NN; k += 256) lds_fj[k + t] = fj_g[b * NN + k + t];
  if (t < 16) lds_fi[t] = fi_g[b * NN + i0 + t];
  __syncthreads();

  const int row = t >> 4;                       // 0..15
  const int sub = t & 15;
  const float fi = lds_fi[row];
  const float* adjrow = adj + (size_t)(i0 + row) * NN;

  // ---- Pass A: per-row max of masked leaky-relu scores -------------------
  float m = -INFINITY;
  for (int k = sub; k < NN; k += 16) {
    const float av = adjrow[k];
    if (av != 0.0f) {
      const float z = fi + lds_fj[k];
      const float e = (z >= 0.f) ? z : NEG_SLOPE * z;
      m = fmaxf(m, e);
    }
  }
#pragma unroll
  for (int off = 1; off < 16; off <<= 1) m = fmaxf(m, __shfl_xor(m, off, 16));

  // ---- Pass B: per-row sum of exp(e - m) ---------------------------------
  float s = 0.f;
  for (int k = sub; k < NN; k += 16) {
    const float av = adjrow[k];
    if (av != 0.0f) {
      const float z = fi + lds_fj[k];
      const float e = (z >= 0.f) ? z : NEG_SLOPE * z;
      s += __expf(e - m);
    }
  }
#pragma unroll
  for (int off = 1; off < 16; off <<= 1) s += __shfl_xor(s, off, 16);
  if (sub == 0) {
    lds_m[row]  = m;
    lds_is[row] = (s > 0.f) ? (1.0f / s) : 0.0f;   // all-masked row -> alpha=0
  }
  __syncthreads();

  // ---- Pass C1: write normalized alpha, fully coalesced ------------------
  float* arow_base = alpha + ((size_t)b * NN + i0) * NN;
  for (int r = 0; r < 16; ++r) {
    const float fir = lds_fi[r], mr = lds_m[r], isr = lds_is[r];
    const float* ar = adj + (size_t)(i0 + r) * NN;
    float* dst = arow_base + (size_t)r * NN;
    for (int k = 0; k < NN; k += 256) {
      const int j = k + t;
      const float av = ar[j];
      float val = 0.f;
      if (av != 0.f) {
        const float z = fir + lds_fj[j];
        const float e = (z >= 0.f) ? z : NEG_SLOPE * z;
        val = __expf(e - mr) * isr;
      }
      dst[j] = val;
    }
  }
  __threadfence();
  __syncthreads();

  // ---- Pass C2: out(16x64) = alpha(16x2048) @ h(2048x64), fp32 WMMA ------
  const int wv   = t >> 5;                      // wave 0..7 -> j slice
  const int lane = t & 31;
  const int mrow = lane & 15;
  const int koff = (lane >> 4) << 1;
  v8f acc[4] = {};
  const float* arowA = arow_base + (size_t)mrow * NN;   // row per lane
  const float* hb    = h + (size_t)b * NN * OUT_DIM;

  const int jend = wv * 256 + 256;
  for (int j = wv * 256; j < jend; j += 4) {
    v2f a;
    a.x = arowA[j + koff];
    a.y = arowA[j + koff + 1];
#pragma unroll
    for (int tt = 0; tt < 4; ++tt) {
      const int n = tt * 16 + mrow;
      v2f bf;
      bf.x = hb[(size_t)(j + koff)     * OUT_DIM + n];
      bf.y = hb[(size_t)(j + koff + 1) * OUT_DIM + n];
      acc[tt] = __builtin_amdgcn_wmma_f32_16x16x4_f32(
          false, a, false, bf, (short)0, acc[tt], false, false);
    }
  }
  // dump wave-partial 16x64 tile to LDS in (m,n) layout
  float* red = lds_red + wv * 16 * OUT_DIM;
#pragma unroll
  for (int tt = 0; tt < 4; ++tt)
#pragma unroll
    for (int r = 0; r < 8; ++r) {
      const int mm = r + ((lane >> 4) << 3);
      red[mm * OUT_DIM + tt * 16 + mrow] = acc[tt][r];
    }
  __syncthreads();

  // reduce 8 wave partials, coalesced store of out tile
  float* outb = out + ((size_t)b * NN + i0) * OUT_DIM;
#pragma unroll
  for (int e4 = 0; e4 < 4; ++e4) {
    const int e = e4 * 256 + t;                 // 0..1023 = 16*64 elems
    float ssum = 0.f;
#pragma unroll
    for (int w = 0; w < 8; ++w) ssum += lds_red[w * 1024 + e];
    outb[e] = ssum;                             // contiguous: i0*64 + e
  }
}

// ---------------------------------------------------------------------------
extern "C" void kernel_launch(void* const* d_in, const int* in_sizes, int n_in,
                              void* d_out, int out_size, void* d_ws, size_t ws_size,
                              hipStream_t stream) {
  (void)in_sizes; (void)n_in; (void)out_size; (void)ws_size;
  const float* x   = (const float*)d_in[0];   // (B, N, 128)
  const float* adj = (const float*)d_in[1];   // (N, N)
  const float* W   = (const float*)d_in[2];   // (128, 64)
  const float* a   = (const float*)d_in[3];   // (128,)

  float* out_p   = (float*)d_out;                                 // (B,N,64)
  float* alpha_p = (float*)d_out + (size_t)BB * NN * OUT_DIM;     // (B,N,N)

  float* h  = (float*)d_ws;                                       // 4 MB
  float* fi = h  + (size_t)BB * NN * OUT_DIM;                     // 64 KB
  float* fj = fi + (size_t)BB * NN;                               // 64 KB

  // h = x @ W : 1024 wave-tiles, 4 waves per block
  gat_h_kernel<<<(BB * NN / 16) / 4, 128, 0, stream>>>(x, W, h);
  // f_i, f_j
  gat_f_kernel<<<(BB * NN) / 256, 256, 0, stream>>>(h, a, fi, fj);
  // softmax + alpha + out
  gat_attn_kernel<<<BB * (NN / 16), 256, 0, stream>>>(adj, h, fi, fj,
                                                      out_p, alpha_p);
}